// EffCapLayer_42193758716027
// MI455X (gfx1250) — compile-verified
//
#include <hip/hip_runtime.h>
#include <cmath>

// EffCapLayer fused kernel for MI455X (gfx1250, wave32).
//
// Design rationale (compile-only; reasoned from hardware):
//  - Reference materializes votes v (256 MB) and streams it ~6x through HBM.
//  - Here: one workgroup = 4 spatial positions; v for those positions (256 KB)
//    lives in LDS (320 KB per WGP on CDNA5), so v never reaches HBM.
//  - The capsule einsum (K=P=4, fp32) maps EXACTLY onto V_WMMA_F32_16X16X4_F32:
//    A = u over 16 rows (4 pos x 4 i), K=j=4, B = W_caps[b] 4x16 (c,k)-tile.
//    fp32 in / fp32 accumulate -> bit-compatible precision with the reference.
//  - Routing (softmax over C, B-reduction, squash, logit update) runs fully in
//    LDS with 64 threads per position.

#define Bc   32   // capsule types in
#define Cc   32   // capsule types out
#define Pp   4
#define P2   16
#define HO   16
#define WO   16
#define NN   16
#define TILE 4                       // spatial positions per workgroup
#define NPOS (NN * HO * WO)          // 4096
#define NWG  (NPOS / TILE)           // 1024

typedef __attribute__((ext_vector_type(2))) float v2f;
typedef __attribute__((ext_vector_type(8))) float v8f;

// LDS partition (floats), dynamic extern shared:
//   v      : TILE*B*C*P2 = 65536   (256 KB)
//   u      : TILE*B*P2   =  2048
//   logits : TILE*B*C    =  4096
//   coup   : TILE*B*C    =  4096
//   out/s  : TILE*C*P2   =  2048
//   a_hat  : TILE*B      =   128
#define OFF_V   0
#define OFF_U   65536
#define OFF_LG  67584
#define OFF_CP  71680
#define OFF_OT  75776
#define OFF_AH  77824
#define LDS_FLOATS 77952            // 311,808 bytes <= 320 KB WGP LDS

__global__ void __launch_bounds__(256, 1)
effcaps_fused_kernel(const float* __restrict__ x,      // (16,32,32,32,16)
                     const float* __restrict__ a,      // (16,32,32,32,1)
                     const float* __restrict__ Wdw,    // (128,1,4,3,3)
                     const float* __restrict__ Wa,     // (32,1,3,3)
                     const float* __restrict__ Wcaps,  // (32,32,4,4)
                     const int*   __restrict__ itersP,
                     float* __restrict__ out)          // p_out(4096*512) ++ a_out(4096*32)
{
    extern __shared__ float smem[];
    float* vS  = smem + OFF_V;    // [pos][b][c][p2]  stride 8192 per pos
    float* uS  = smem + OFF_U;    // [pos][b][p2]     stride 512
    float* lgS = smem + OFF_LG;   // [pos][b][c]      stride 1024
    float* cpS = smem + OFF_CP;   // [pos][b][c]      stride 1024
    float* otS = smem + OFF_OT;   // [pos][c][p2]     stride 512
    float* ahS = smem + OFF_AH;   // [pos][b]         stride 32

    const int tid = threadIdx.x;
    const int wg  = blockIdx.x;
    const int iters = itersP[0];

    // ---------------- Phase A: depthwise conv -> u poses, and a_hat ----------
    // u[pos][b][i*4+dd] = sum_{kd,kh,kw} Wdw[b*4+i][kd][kh][kw] *
    //                     x[n][2hi+kh-1][2wi+kw-1][b][dd*4+kd]
    #pragma unroll
    for (int e = 0; e < 8; ++e) {
        const int idx = tid + e * 256;           // 2048 u entries
        const int pos = idx >> 9;
        const int rem = idx & 511;
        const int b   = rem >> 4;
        const int p2  = rem & 15;
        const int i   = p2 >> 2;
        const int dd  = p2 & 3;
        const int pid = wg * TILE + pos;
        const int nI  = pid >> 8;
        const int hi  = (pid >> 4) & 15;
        const int wi  = pid & 15;
        const float* wp = Wdw + (size_t)(b * 4 + i) * 36;
        float acc = 0.f;
        for (int kh = 0; kh < 3; ++kh) {
            const int hin = 2 * hi + kh - 1;
            if (hin < 0 || hin >= 32) continue;
            for (int kw = 0; kw < 3; ++kw) {
                const int win = 2 * wi + kw - 1;
                if (win < 0 || win >= 32) continue;
                const float* xp = x + ((((size_t)nI * 32 + hin) * 32 + win) * 32 + b) * 16 + dd * 4;
                #pragma unroll
                for (int kd = 0; kd < 4; ++kd)
                    acc = fmaf(wp[kd * 9 + kh * 3 + kw], xp[kd], acc);
            }
        }
        uS[pos * 512 + b * 16 + p2] = acc;
    }
    // a_hat[pos][b] = relu(3x3 depthwise conv of a)
    if (tid < TILE * Bc) {
        const int pos = tid >> 5;
        const int b   = tid & 31;
        const int pid = wg * TILE + pos;
        const int nI  = pid >> 8;
        const int hi  = (pid >> 4) & 15;
        const int wi  = pid & 15;
        float acc = 0.f;
        for (int kh = 0; kh < 3; ++kh) {
            const int hin = 2 * hi + kh - 1;
            if (hin < 0 || hin >= 32) continue;
            for (int kw = 0; kw < 3; ++kw) {
                const int win = 2 * wi + kw - 1;
                if (win < 0 || win >= 32) continue;
                acc = fmaf(Wa[b * 9 + kh * 3 + kw],
                           a[(((size_t)nI * 32 + hin) * 32 + win) * 32 + b], acc);
            }
        }
        ahS[pos * 32 + b] = fmaxf(acc, 0.f);
    }
    __syncthreads();

    // ---------------- Phase B: votes via V_WMMA_F32_16X16X4_F32 --------------
    // Per b: D(16x16) = A(16x4) x B(4x16); M rows = (pos,i), K = j, N = (c,k).
    {
        const int wave  = tid >> 5;
        const int lane  = tid & 31;
        const int m     = lane & 15;        // A row / D column index group
        const int khalf = lane >> 4;        // K-half for A/B, M-half for D
        const int j0    = khalf * 2;
        const int posA  = m >> 2;
        const int iA    = m & 3;
        const int cSub  = m >> 2;           // N col -> (c within tile, k)
        const int kk    = m & 3;
        #pragma unroll
        for (int bb = 0; bb < 4; ++bb) {
            const int b = wave * 4 + bb;
            v2f Aop;
            Aop.x = uS[posA * 512 + b * 16 + iA * 4 + j0];
            Aop.y = uS[posA * 512 + b * 16 + iA * 4 + j0 + 1];
            const float* Wb = Wcaps + (size_t)b * 512;
            #pragma unroll
            for (int ct = 0; ct < 8; ++ct) {
                const int c = ct * 4 + cSub;
                v2f Bop;
                Bop.x = Wb[c * 16 + j0 * 4 + kk];
                Bop.y = Wb[c * 16 + (j0 + 1) * 4 + kk];
                v8f acc = {};
                acc = __builtin_amdgcn_wmma_f32_16x16x4_f32(
                        false, Aop, false, Bop, (short)0, acc, false, false);
                #pragma unroll
                for (int r = 0; r < 8; ++r) {
                    const int M    = r + khalf * 8;   // D row
                    const int posD = M >> 2;
                    const int iD   = M & 3;
                    vS[posD * 8192 + b * 512 + c * 16 + iD * 4 + kk] = acc[r];
                }
            }
        }
    }
    __syncthreads();

    // ---------------- Phase C: dynamic routing (64 threads per position) -----
    {
        const int sub = tid & 63;
        const int pos = tid >> 6;
        float* vP = vS  + pos * 8192;
        float* lg = lgS + pos * 1024;
        float* cp = cpS + pos * 1024;
        float* ot = otS + pos * 512;
        float* ah = ahS + pos * 32;

        for (int idx = sub; idx < 1024; idx += 64) lg[idx] = 0.f;
        __syncthreads();

        for (int it = 0; it < iters; ++it) {
            // coupling: softmax over c per b, scaled by a_hat[b]
            if (sub < Bc) {
                const int b = sub;
                float mx = -3.4e38f;
                for (int c = 0; c < Cc; ++c) mx = fmaxf(mx, lg[b * 32 + c]);
                float se = 0.f;
                for (int c = 0; c < Cc; ++c) se += __expf(lg[b * 32 + c] - mx);
                const float scl = ah[b] / se;
                for (int c = 0; c < Cc; ++c)
                    cp[b * 32 + c] = __expf(lg[b * 32 + c] - mx) * scl;
            }
            __syncthreads();
            // s[c][p2] = sum_b coup[b][c] * v[b][c][p2]
            for (int idx = sub; idx < Cc * P2; idx += 64) {
                const int c  = idx >> 4;
                const int p2 = idx & 15;
                float s = 0.f;
                for (int b = 0; b < Bc; ++b)
                    s = fmaf(cp[b * 32 + c], vP[b * 512 + c * 16 + p2], s);
                ot[idx] = s;
            }
            __syncthreads();
            // squash per c
            if (sub < Cc) {
                const int c = sub;
                float n2 = 0.f;
                for (int p2 = 0; p2 < P2; ++p2) {
                    const float sv = ot[c * 16 + p2];
                    n2 = fmaf(sv, sv, n2);
                }
                const float scale = n2 / (1.f + n2) * rsqrtf(n2 + 1e-8f);
                for (int p2 = 0; p2 < P2; ++p2) ot[c * 16 + p2] *= scale;
            }
            __syncthreads();
            // logits[b][c] += <v[b][c][:], out[c][:]>   (not needed after last)
            if (it + 1 < iters) {
                for (int idx = sub; idx < Bc * Cc; idx += 64) {
                    const int b = idx >> 5;
                    const int c = idx & 31;
                    float d = 0.f;
                    for (int p2 = 0; p2 < P2; ++p2)
                        d = fmaf(vP[b * 512 + c * 16 + p2], ot[c * 16 + p2], d);
                    lg[idx] += d;
                }
                __syncthreads();
            }
        }

        // ---------------- outputs ----------------
        const int pid = wg * TILE + pos;
        const size_t pbase = (size_t)pid * (Cc * P2);
        for (int idx = sub; idx < Cc * P2; idx += 64)
            out[pbase + idx] = ot[idx];
        if (sub < Cc) {
            const int c = sub;
            float n2 = 0.f;
            for (int p2 = 0; p2 < P2; ++p2) {
                const float ov = ot[c * 16 + p2];
                n2 = fmaf(ov, ov, n2);
            }
            out[(size_t)NPOS * (Cc * P2) + (size_t)pid * Cc + c] = sqrtf(n2 + 1e-8f);
        }
    }
}

extern "C" void kernel_launch(void* const* d_in, const int* in_sizes, int n_in,
                              void* d_out, int out_size, void* d_ws, size_t ws_size,
                              hipStream_t stream) {
    const float* x     = (const float*)d_in[0];
    const float* a     = (const float*)d_in[1];
    const float* Wdw   = (const float*)d_in[2];
    const float* Wa    = (const float*)d_in[3];
    const float* Wcaps = (const float*)d_in[4];
    const int*   iters = (const int*)d_in[5];
    float* out = (float*)d_out;
    (void)in_sizes; (void)n_in; (void)out_size; (void)d_ws; (void)ws_size;

    const size_t shmem = (size_t)LDS_FLOATS * sizeof(float); // ~305 KB (<=320 KB/WGP)
    hipFuncSetAttribute((const void*)effcaps_fused_kernel,
                        hipFuncAttributeMaxDynamicSharedMemorySize, (int)shmem);
    effcaps_fused_kernel<<<NWG, 256, shmem, stream>>>(x, a, Wdw, Wa, Wcaps, iters, out);
}